// ContextAttentionBlock_65317862637877
// MI455X (gfx1250) — compile-verified
//
#include <hip/hip_runtime.h>
#include <math.h>

// Problem constants (from reference setup_inputs)
#define BB   4
#define LSEQ 1024
#define EE   512
#define HH   8
#define DH   64
#define TAU  32

typedef __attribute__((ext_vector_type(2))) float v2f;
typedef __attribute__((ext_vector_type(8))) float v8f;

// CDNA5 fp32 WMMA: D(16x16,f32) = A(16x4,f32) x B(4x16,f32) + C
__device__ __forceinline__ v8f wmma4(v2f a, v2f b, v8f c) {
    return __builtin_amdgcn_wmma_f32_16x16x4_f32(
        false, a, false, b, (short)0, c, false, false);
}

// Async global->LDS copy (16 bytes per lane), tracked by ASYNCcnt.
// VDST = LDS byte address (generic LDS addr truncates to its low 32 bits).
__device__ __forceinline__ void async_copy_b128(void* lds_dst, const float* gsrc) {
    unsigned lds_addr = (unsigned)(size_t)lds_dst;
    asm volatile("global_load_async_to_lds_b128 %0, %1, off"
                 :: "v"(lds_addr), "v"(gsrc) : "memory");
}
__device__ __forceinline__ void async_wait0() {
    asm volatile("s_wait_asynccnt 0" ::: "memory");
}

// ---------------------------------------------------------------------------
// Kernel 1: batched projection GEMM.  dst = src @ W + bias
// src: (4096 x 512) row-major, W: (512 x 512) row-major, bias: (512)
// grid = (M/64, N/64, 7), block = 128 (4 waves), wave -> 32x32 sub-tile
// ---------------------------------------------------------------------------
struct GemmDesc { const float* src; const float* W; const float* bias; float* dst; };
struct GemmArgs { GemmDesc g[7]; };

__global__ __launch_bounds__(128) void proj_gemm_kernel(GemmArgs args) {
    const GemmDesc gd = args.g[blockIdx.z];
    const int row0 = blockIdx.x * 64;
    const int col0 = blockIdx.y * 64;

    __shared__ __align__(16) float As[64][16];
    __shared__ __align__(16) float Bs[16][64];

    const int tid  = threadIdx.x;
    const int lane = tid & 31;
    const int w    = tid >> 5;
    const int wm   = (w >> 1) * 32;
    const int wn   = (w & 1) * 32;
    const int n16  = lane & 15;          // m index (A) / n index (B,C)
    const int khi  = (lane >> 4) * 2;    // k sub-offset for A/B frags
    const int mb   = (lane >> 4) * 8;    // m base for C/D rows

    v8f acc[2][2] = {};

    for (int kb = 0; kb < EE; kb += 16) {
        // Stage A tile 64x16 and B tile 16x64 via async global->LDS copies.
        for (int idx = tid; idx < 64 * 4; idx += 128) {
            int r = idx >> 2, c4 = (idx & 3) * 4;
            async_copy_b128(&As[r][c4],
                            &gd.src[(size_t)(row0 + r) * EE + kb + c4]);
        }
        for (int idx = tid; idx < 16 * 16; idx += 128) {
            int r = idx >> 4, c4 = (idx & 15) * 4;
            async_copy_b128(&Bs[r][c4],
                            &gd.W[(size_t)(kb + r) * EE + col0 + c4]);
        }
        async_wait0();
        __syncthreads();

#pragma unroll
        for (int kc = 0; kc < 4; ++kc) {
            v2f a0, a1, b0, b1;
            a0.x = As[wm + n16][kc * 4 + khi];
            a0.y = As[wm + n16][kc * 4 + khi + 1];
            a1.x = As[wm + 16 + n16][kc * 4 + khi];
            a1.y = As[wm + 16 + n16][kc * 4 + khi + 1];
            b0.x = Bs[kc * 4 + khi][wn + n16];
            b0.y = Bs[kc * 4 + khi + 1][wn + n16];
            b1.x = Bs[kc * 4 + khi][wn + 16 + n16];
            b1.y = Bs[kc * 4 + khi + 1][wn + 16 + n16];
            acc[0][0] = wmma4(a0, b0, acc[0][0]);
            acc[0][1] = wmma4(a0, b1, acc[0][1]);
            acc[1][0] = wmma4(a1, b0, acc[1][0]);
            acc[1][1] = wmma4(a1, b1, acc[1][1]);
        }
        __syncthreads();
    }

#pragma unroll
    for (int mt = 0; mt < 2; ++mt)
#pragma unroll
        for (int nt = 0; nt < 2; ++nt) {
            const int gcol = col0 + wn + nt * 16 + n16;
            const float bias = gd.bias[gcol];
#pragma unroll
            for (int r = 0; r < 8; ++r) {
                const int grow = row0 + wm + mt * 16 + mb + r;
                gd.dst[(size_t)grow * EE + gcol] = acc[mt][nt][r] + bias;
            }
        }
}

// ---------------------------------------------------------------------------
// Kernel 2: sliding-window attention (tau=32) on projected tensors.
// For row l, head h: scores over K rows l-32..l-1 (OOB rows == bias vector),
// softmax, weighted sum of V rows.  grid = (L/64, H, B), block = 128.
// Wave w handles 16 rows; its window union is LDS rows [16w, 16w+47].
// ---------------------------------------------------------------------------
__global__ __launch_bounds__(128) void local_window_attn_kernel(
    const float* __restrict__ Qp, const float* __restrict__ Kp,
    const float* __restrict__ Vp, const float* __restrict__ bvec,
    float* __restrict__ outp) {
    const int l0 = blockIdx.x * 64;
    const int h  = blockIdx.y;
    const int b  = blockIdx.z;

    __shared__ __align__(16) float Kw[96][DH];
    __shared__ __align__(16) float Vw[96][DH];
    __shared__ __align__(16) float Pm[4][16][48];

    const int tid  = threadIdx.x;
    const int lane = tid & 31;
    const int w    = tid >> 5;
    const int n16  = lane & 15;
    const int khi  = (lane >> 4) * 2;
    const int mb   = (lane >> 4) * 8;
    const size_t hoff = (size_t)h * DH;
    const float scale = 0.125f;  // d^-0.5, d=64

    // Stage 96 window rows (l0-32 .. l0+63) of projected K and V (head slice).
    for (int idx = tid; idx < 96 * 16; idx += 128) {
        int jr = idx >> 4, c4 = (idx & 15) * 4;
        int jg = l0 - TAU + jr;
        float4 kv, vv;
        if (jg >= 0) {
            size_t base = ((size_t)b * LSEQ + jg) * EE + hoff;
            kv = *(const float4*)&Kp[base + c4];
            vv = *(const float4*)&Vp[base + c4];
        } else {
            kv = *(const float4*)&bvec[hoff + c4];  // zero-row @ W + b == b
            vv = kv;
        }
        *(float4*)&Kw[jr][c4] = kv;
        *(float4*)&Vw[jr][c4] = vv;
    }
    __syncthreads();

    const int lw = l0 + 16 * w;

    // S(16x48) = Q(16x64) @ Kwin^T
    v8f sacc[3] = {};
    const size_t qrow = ((size_t)b * LSEQ + lw + n16) * EE + hoff;
#pragma unroll 4
    for (int kc = 0; kc < 16; ++kc) {
        v2f a;
        a.x = Qp[qrow + kc * 4 + khi];
        a.y = Qp[qrow + kc * 4 + khi + 1];
#pragma unroll
        for (int nt = 0; nt < 3; ++nt) {
            v2f bf;
            const int r = 16 * w + nt * 16 + n16;
            bf.x = Kw[r][kc * 4 + khi];
            bf.y = Kw[r][kc * 4 + khi + 1];
            sacc[nt] = wmma4(a, bf, sacc[nt]);
        }
    }
#pragma unroll
    for (int nt = 0; nt < 3; ++nt)
#pragma unroll
        for (int r = 0; r < 8; ++r)
            Pm[w][mb + r][nt * 16 + n16] = sacc[nt][r];
    __syncthreads();

    // Masked row softmax: row m valid for jj in [m, m+32)
    if (lane < 16) {
        const int m = lane;
        float mx = -3.4e38f;
        for (int jj = m; jj < m + TAU; ++jj)
            mx = fmaxf(mx, Pm[w][m][jj] * scale);
        float s = 0.f;
        for (int jj = 0; jj < 48; ++jj) {
            float p = 0.f;
            if (jj >= m && jj < m + TAU) {
                p = __expf(Pm[w][m][jj] * scale - mx);
                s += p;
            }
            Pm[w][m][jj] = p;
        }
        const float inv = 1.f / s;
        for (int jj = m; jj < m + TAU; ++jj) Pm[w][m][jj] *= inv;
    }
    __syncthreads();

    // O(16x64) = P(16x48) @ Vwin(48x64)
    v8f oacc[4] = {};
#pragma unroll 4
    for (int kc = 0; kc < 12; ++kc) {
        v2f a;
        a.x = Pm[w][n16][kc * 4 + khi];
        a.y = Pm[w][n16][kc * 4 + khi + 1];
#pragma unroll
        for (int nt = 0; nt < 4; ++nt) {
            v2f bf;
            bf.x = Vw[16 * w + kc * 4 + khi][nt * 16 + n16];
            bf.y = Vw[16 * w + kc * 4 + khi + 1][nt * 16 + n16];
            oacc[nt] = wmma4(a, bf, oacc[nt]);
        }
    }
#pragma unroll
    for (int nt = 0; nt < 4; ++nt)
#pragma unroll
        for (int r = 0; r < 8; ++r)
            outp[((size_t)b * LSEQ + lw + mb + r) * EE + hoff + nt * 16 + n16] =
                oacc[nt][r];
}

// ---------------------------------------------------------------------------
// Kernel 3: full attention with online (flash) softmax over L=1024 in 64-col
// chunks.  grid = (L/64, H, B), block = 128; wave w owns 16 query rows.
// K/V chunk staging uses async global->LDS copies (ASYNCcnt).
// ---------------------------------------------------------------------------
__global__ __launch_bounds__(128) void global_attn_kernel(
    const float* __restrict__ qf, const float* __restrict__ kf,
    const float* __restrict__ vf, float* __restrict__ outp) {
    const int l0 = blockIdx.x * 64;
    const int h  = blockIdx.y;
    const int b  = blockIdx.z;

    __shared__ __align__(16) float Ks[64][DH];
    __shared__ __align__(16) float Vs[64][DH];
    __shared__ __align__(16) float Pm[4][16][64];
    __shared__ float rowf[4][16];

    const int tid  = threadIdx.x;
    const int lane = tid & 31;
    const int w    = tid >> 5;
    const int n16  = lane & 15;
    const int khi  = (lane >> 4) * 2;
    const int mb   = (lane >> 4) * 8;
    const size_t hoff = (size_t)h * DH;
    const int lw = l0 + 16 * w;
    const float scale = 0.125f;

    // Preload this wave's Q fragments (16 k-chunks of 4)
    v2f aq[16];
    const size_t qrow = ((size_t)b * LSEQ + lw + n16) * EE + hoff;
#pragma unroll
    for (int kc = 0; kc < 16; ++kc) {
        aq[kc].x = qf[qrow + kc * 4 + khi];
        aq[kc].y = qf[qrow + kc * 4 + khi + 1];
    }

    v8f oacc[4] = {};
    float runmax = -3.4e38f, runsum = 0.f;  // valid in lanes 0..15

    for (int c = 0; c < 16; ++c) {
        const int m0 = c * 64;
        for (int idx = tid; idx < 64 * 16; idx += 128) {
            int r = idx >> 4, c4 = (idx & 15) * 4;
            size_t base = ((size_t)b * LSEQ + m0 + r) * EE + hoff + c4;
            async_copy_b128(&Ks[r][c4], &kf[base]);
            async_copy_b128(&Vs[r][c4], &vf[base]);
        }
        async_wait0();
        __syncthreads();

        // S(16x64) = Q @ Kc^T
        v8f sacc[4] = {};
#pragma unroll 4
        for (int kc = 0; kc < 16; ++kc) {
#pragma unroll
            for (int nt = 0; nt < 4; ++nt) {
                v2f bf;
                bf.x = Ks[nt * 16 + n16][kc * 4 + khi];
                bf.y = Ks[nt * 16 + n16][kc * 4 + khi + 1];
                sacc[nt] = wmma4(aq[kc], bf, sacc[nt]);
            }
        }
#pragma unroll
        for (int nt = 0; nt < 4; ++nt)
#pragma unroll
            for (int r = 0; r < 8; ++r)
                Pm[w][mb + r][nt * 16 + n16] = sacc[nt][r];
        __syncthreads();

        // Online softmax update (one row per lane 0..15)
        if (lane < 16) {
            const int m = lane;
            float cmx = -3.4e38f;
            for (int jj = 0; jj < 64; ++jj)
                cmx = fmaxf(cmx, Pm[w][m][jj] * scale);
            const float nmx  = fmaxf(runmax, cmx);
            const float corr = __expf(runmax - nmx);
            float s = 0.f;
            for (int jj = 0; jj < 64; ++jj) {
                float p = __expf(Pm[w][m][jj] * scale - nmx);
                s += p;
                Pm[w][m][jj] = p;
            }
            runsum = runsum * corr + s;
            runmax = nmx;
            rowf[w][m] = corr;
        }
        __syncthreads();

        // Rescale accumulator rows, then O += P @ Vc
#pragma unroll
        for (int r = 0; r < 8; ++r) {
            const float f = rowf[w][mb + r];
#pragma unroll
            for (int nt = 0; nt < 4; ++nt) oacc[nt][r] *= f;
        }
#pragma unroll 4
        for (int kc = 0; kc < 16; ++kc) {
            v2f a;
            a.x = Pm[w][n16][kc * 4 + khi];
            a.y = Pm[w][n16][kc * 4 + khi + 1];
#pragma unroll
            for (int nt = 0; nt < 4; ++nt) {
                v2f bf;
                bf.x = Vs[kc * 4 + khi][nt * 16 + n16];
                bf.y = Vs[kc * 4 + khi + 1][nt * 16 + n16];
                oacc[nt] = wmma4(a, bf, oacc[nt]);
            }
        }
        __syncthreads();
    }

    if (lane < 16) rowf[w][lane] = 1.f / runsum;
    __syncthreads();
#pragma unroll
    for (int nt = 0; nt < 4; ++nt)
#pragma unroll
        for (int r = 0; r < 8; ++r) {
            const float f = rowf[w][mb + r];
            outp[((size_t)b * LSEQ + lw + mb + r) * EE + hoff + nt * 16 + n16] =
                oacc[nt][r] * f;
        }
}

// ---------------------------------------------------------------------------
extern "C" void kernel_launch(void* const* d_in, const int* in_sizes, int n_in,
                              void* d_out, int out_size, void* d_ws, size_t ws_size,
                              hipStream_t stream) {
    (void)in_sizes; (void)n_in; (void)out_size; (void)ws_size;
    const float* query = (const float*)d_in[0];
    const float* key   = (const float*)d_in[1];
    const float* value = (const float*)d_in[2];
    const float* Wq    = (const float*)d_in[3];
    const float* bq    = (const float*)d_in[4];
    const float* Wk    = (const float*)d_in[5];
    const float* bk    = (const float*)d_in[6];
    const float* Wv    = (const float*)d_in[7];
    const float* bv    = (const float*)d_in[8];
    float* out = (float*)d_out;
    float* ws  = (float*)d_ws;

    const size_t S1 = (size_t)BB * LSEQ * EE;  // 2M floats per buffer
    float* Aq = ws + 0 * S1;  // query @ Wq + bq   (q-role of q-block)
    float* Kq = ws + 1 * S1;  // key   @ Wq + bq   (k-role of q-block)
    float* Vq = ws + 2 * S1;  // value @ Wq + bq   (v-role of q-block)
    float* Ak = ws + 3 * S1;  // key   @ Wk + bk   (q-role of k-block)
    float* Qk = ws + 4 * S1;  // query @ Wk + bk   (k-role of k-block)
    float* Vk = ws + 5 * S1;  // value @ Wk + bk   (v-role of k-block)
    float* Vv = ws + 6 * S1;  // value @ Wv + bv   (final V)
    float* qf = ws + 7 * S1;  // q_final
    float* kf = ws + 8 * S1;  // k_final

    GemmArgs ga;
    ga.g[0] = GemmDesc{query, Wq, bq, Aq};
    ga.g[1] = GemmDesc{key,   Wq, bq, Kq};
    ga.g[2] = GemmDesc{value, Wq, bq, Vq};
    ga.g[3] = GemmDesc{key,   Wk, bk, Ak};
    ga.g[4] = GemmDesc{query, Wk, bk, Qk};
    ga.g[5] = GemmDesc{value, Wk, bk, Vk};
    ga.g[6] = GemmDesc{value, Wv, bv, Vv};

    proj_gemm_kernel<<<dim3(BB * LSEQ / 64, EE / 64, 7), 128, 0, stream>>>(ga);
    local_window_attn_kernel<<<dim3(LSEQ / 64, HH, BB), 128, 0, stream>>>(
        Aq, Kq, Vq, bq, qf);
    local_window_attn_kernel<<<dim3(LSEQ / 64, HH, BB), 128, 0, stream>>>(
        Ak, Qk, Vk, bk, kf);
    global_attn_kernel<<<dim3(LSEQ / 64, HH, BB), 128, 0, stream>>>(
        qf, kf, Vv, out);
}